// WaveletTransform_77395310674219
// MI455X (gfx1250) — compile-verified
//
#include <hip/hip_runtime.h>

typedef float v2f __attribute__((ext_vector_type(2)));
typedef float v4f __attribute__((ext_vector_type(4)));
typedef float v8f __attribute__((ext_vector_type(8)));

// x: [B=4, C=32, H=512, W=512] f32 -> (cA,cH,cV,cD) each [4,32,256,256] f32,
// concatenated flat in d_out in that order.
static constexpr int    kH = 512, kW = 512, kOH = 256, kOW = 256;
static constexpr int    kPlanes       = 4 * 32;                       // 128
static constexpr size_t kSubbandElems = (size_t)kPlanes * kOH * kOW;  // 8388608

__global__ __launch_bounds__(256) void haar_dwt2_wmma(const float* __restrict__ x,
                                                      float* __restrict__ out) {
  const int tid  = threadIdx.x;
  const int lane = tid & 31;
  const int wave = tid >> 5;     // 8 waves / workgroup, one output row each
  const int rp   = lane >> 4;    // row parity: lanes 0-15 even row, 16-31 odd row
  const int ci   = lane & 15;    // column sub-index within tile

  const int plane = blockIdx.x >> 5;     // 0..127  (b*32 + c)
  const int rg    = blockIdx.x & 31;     // 0..31   row group
  const int orow  = rg * 8 + wave;       // 0..255  output row

  // A-operand, 16x4 f32 Haar matrix (constant): row m holds coeff set (m&3)
  //   r=0:cA(+ + + +)  r=1:cH(+ + - -)  r=2:cV(+ - + -)  r=3:cD(+ - - +), x0.5
  // ISA A layout: lanes 0-15 hold K=0 (v0), K=1 (v1); lanes 16-31 hold K=2,3.
  const int r  = ci & 3;
  const int k0 = rp * 2;
  const int k1 = k0 + 1;
  const int n0 = ((r & 1) & (k0 >> 1)) ^ (((r >> 1) & 1) & (k0 & 1));
  const int n1 = ((r & 1) & (k1 >> 1)) ^ (((r >> 1) & 1) & (k1 & 1));
  v2f amat;
  amat.x = n0 ? -0.5f : 0.5f;
  amat.y = n1 ? -0.5f : 0.5f;

  // Each lane streams float4's from input row 2*orow+rp; lanes 0-15 cover the
  // even row (256 B/iter), lanes 16-31 the odd row -> one b128 load per iter.
  const float* src = x + (size_t)plane * (kH * kW)
                       + (size_t)(2 * orow + rp) * kW + 4 * ci;

  // Merged-store pattern: lanes 0-15 write even output cols (from even-block
  // WMMA), lanes 16-31 write odd output cols (from odd-block WMMA). The 32
  // addresses permute a contiguous 128 B range -> fully coalesced.
  const size_t obase = (size_t)plane * (kOH * kOW) + (size_t)orow * kOW
                     + 2 * ci + rp;
  float* oA = out + 0 * kSubbandElems + obase;
  float* oH = out + 1 * kSubbandElems + obase;
  float* oV = out + 2 * kSubbandElems + obase;
  float* oD = out + 3 * kSubbandElems + obase;

  const v8f cz = {};
#pragma unroll
  for (int t = 0; t < 8; ++t) {
    // 512 B per wave: 64 input cols x 2 rows = 32 Haar blocks. NT policy:
    // 256 MiB stream > 192 MB L2, keep it out of the cache.
    v4f f = __builtin_nontemporal_load((const v4f*)(src + t * 64));

    // B-operand layout (4x16): lanes 0-15 K=0,1 ; lanes 16-31 K=2,3.
    // (f.x,f.y) == (x00,x01)|(x10,x11) for even blocks; (f.z,f.w) for odd.
    v2f b0; b0.x = f.x; b0.y = f.y;
    v2f b1; b1.x = f.z; b1.y = f.w;

    v8f d0 = __builtin_amdgcn_wmma_f32_16x16x4_f32(false, amat, false, b0,
                                                   (short)0, cz, false, false);
    v8f d1 = __builtin_amdgcn_wmma_f32_16x16x4_f32(false, amat, false, b1,
                                                   (short)0, cz, false, false);

    // D vgpr c = coeff c: lanes 0-15 valid in d0 (even blocks); lanes 16-31 of
    // d1 carry the row-(c+8) duplicate = coeff c of the odd blocks.
    float vA = rp ? d1[0] : d0[0];
    float vH = rp ? d1[1] : d0[1];
    float vV = rp ? d1[2] : d0[2];
    float vD = rp ? d1[3] : d0[3];

    __builtin_nontemporal_store(vA, oA + t * 32);
    __builtin_nontemporal_store(vH, oH + t * 32);
    __builtin_nontemporal_store(vV, oV + t * 32);
    __builtin_nontemporal_store(vD, oD + t * 32);
  }
}

extern "C" void kernel_launch(void* const* d_in, const int* /*in_sizes*/, int /*n_in*/,
                              void* d_out, int /*out_size*/, void* /*d_ws*/, size_t /*ws_size*/,
                              hipStream_t stream) {
  const float* x   = (const float*)d_in[0];
  float*       out = (float*)d_out;
  // 128 planes x 32 row-groups; 256 threads = 8 wave32 per workgroup.
  haar_dwt2_wmma<<<dim3(kPlanes * 32), dim3(256), 0, stream>>>(x, out);
}